// STCA_Loss_80504866996731
// MI455X (gfx1250) — compile-verified
//
#include <hip/hip_runtime.h>
#include <hip/hip_bf16.h>

// STCA loss on MI455X (gfx1250).
// Memory-bound streaming scan: 256 MiB of vmem read once -> ~11us floor at
// 23.3 TB/s. No matrix contraction exists in this workload, so no WMMA; the
// CDNA5-specific data path used is GLOBAL_LOAD_ASYNC_TO_LDS_B128 (ASYNCcnt)
// staging each 8KB trace into LDS, then all cluster logic runs out of LDS.

#define T_LEN     2048
#define C_GAP     3
#define KMAX      512          // (T + C) / (C + 1)
#define NTHR      256
#define PERTHR    8            // T_LEN / NTHR

#if defined(__HIP_DEVICE_COMPILE__) && defined(__gfx1250__) && \
    __has_builtin(__builtin_amdgcn_global_load_async_to_lds_b128) && \
    __has_builtin(__builtin_amdgcn_s_wait_asynccnt)
#define USE_ASYNC_LDS 1
#else
#define USE_ASYNC_LDS 0
#endif

#if USE_ASYNC_LDS
typedef int v4i __attribute__((ext_vector_type(4)));
typedef __attribute__((address_space(1))) v4i* gvec_t;   // global int4*
typedef __attribute__((address_space(3))) v4i* lvec_t;   // LDS int4*
#endif

__global__ __launch_bounds__(NTHR) void stca_trace_kernel(
    const float* __restrict__ vmem,
    const int*   __restrict__ labels,
    float*       __restrict__ out_spk,   // [B*N] cluster counts as float
    float*       __restrict__ contrib,   // [B*N] per-trace loss term (or null)
    float*       __restrict__ loss_atomic,
    int Ndim)
{
    const int r   = blockIdx.x;          // trace id = b*N + n
    const int b   = r / Ndim;
    const int n   = r - b * Ndim;
    const int tid = threadIdx.x;

    __shared__ float         sv[T_LEN];        // 8 KB trace
    __shared__ unsigned char sbyte[NTHR];      // per-thread 8-bit spike mask
    __shared__ int           sscan[NTHR];
    __shared__ int           s_cnt[KMAX];
    __shared__ int           s_first[KMAX];
    __shared__ int           s_last[KMAX];
    __shared__ int           s_key;
    __shared__ float         sred_f[NTHR];
    __shared__ int           sred_i[NTHR];

    const float* g = vmem + (size_t)r * T_LEN;

    // ---- Stage trace into LDS --------------------------------------------
#if USE_ASYNC_LDS
    {
        float* gnc = const_cast<float*>(g);
        gvec_t gp = (gvec_t)gnc;
        lvec_t lp = (lvec_t)(float*)sv;
        // 32 bytes per lane = 2 x b128; 256 lanes cover the whole 8KB trace.
        __builtin_amdgcn_global_load_async_to_lds_b128(gp + tid * 2,     lp + tid * 2,     0, 0);
        __builtin_amdgcn_global_load_async_to_lds_b128(gp + tid * 2 + 1, lp + tid * 2 + 1, 0, 0);
    }
#else
    {
        const float4* g4 = (const float4*)g;
        float4* s4 = (float4*)sv;
        for (int i = tid; i < T_LEN / 4; i += NTHR) s4[i] = g4[i];
    }
#endif

    // init cluster stat arrays while the DMA is in flight
    for (int k = tid; k < KMAX; k += NTHR) {
        s_cnt[k]   = 0;
        s_first[k] = 0x7fffffff;
        s_last[k]  = -1;
    }
    if (tid == 0) s_key = 0x7fffffff;

#if USE_ASYNC_LDS
    __builtin_amdgcn_s_wait_asynccnt(0);
#endif
    __syncthreads();

    // ---- Spike mask for this thread's 8 timesteps ------------------------
    const int t0 = tid * PERTHR;
    unsigned m = 0u;
    float vmax_local = -3.402823466e38f;
    float vloc[PERTHR];
#pragma unroll
    for (int j = 0; j < PERTHR; ++j) {
        float v = sv[t0 + j];
        vloc[j] = v;
        if (v >= 0.0f) m |= (1u << j);
        vmax_local = fmaxf(vmax_local, v);
    }
    sbyte[tid] = (unsigned char)m;
    __syncthreads();

    // start[t] = spike[t] && no spike in [t-3, t-1]  (bit trick on 11-bit window)
    unsigned prev = (tid > 0) ? (unsigned)sbyte[tid - 1] : 0u;
    unsigned w = (m << 3) | (prev >> (8 - C_GAP));   // bits 0..2 = spikes t-3..t-1
    unsigned q = (w | (w >> 1) | (w >> 2));          // q[j] = any spike in [t-3,t-1]
    unsigned start = m & ~q & 0xFFu;
    int pc = __popc(start);

    // ---- Exclusive scan of start counts (Hillis-Steele over 256) ---------
    sscan[tid] = pc;
    __syncthreads();
    for (int off = 1; off < NTHR; off <<= 1) {
        int v = (tid >= off) ? sscan[tid - off] : 0;
        __syncthreads();
        sscan[tid] += v;
        __syncthreads();
    }
    const int base        = sscan[tid] - pc;     // clusters started before chunk
    const int num_cluster = sscan[NTHR - 1];

    // ---- Per-cluster count/first/last via LDS integer atomics ------------
    int cid = base - 1;
#pragma unroll
    for (int j = 0; j < PERTHR; ++j) {
        if ((start >> j) & 1u) cid++;
        if ((m >> j) & 1u) {
            int c = cid < 0 ? 0 : cid;           // provably >=0; clamp for safety
            int t = t0 + j;
            atomicAdd(&s_cnt[c], 1);
            atomicMin(&s_first[c], t);
            atomicMax(&s_last[c], t);
        }
    }
    __syncthreads();

    // ---- argmin over counts, ties -> smallest k (packed key) -------------
    for (int k = tid; k < num_cluster; k += NTHR) {
        atomicMin(&s_key, (s_cnt[k] << 10) | k); // cnt<=2048, k<512 -> fits
    }
    __syncthreads();

    // ---- Mean of positive v over smallest cluster's [first,last] span ----
    float psum = 0.0f;
    int   pcnt = 0;
    if (num_cluster > 0) {
        int kmin = s_key & 1023;
        int fmin = s_first[kmin];
        int lmin = s_last[kmin];
#pragma unroll
        for (int j = 0; j < PERTHR; ++j) {
            int t = t0 + j;
            float v = vloc[j];
            if (t >= fmin && t <= lmin && v > 0.0f) { psum += v; pcnt++; }
        }
    }

    // sum / count reduction (deterministic tree)
    sred_f[tid] = psum;
    sred_i[tid] = pcnt;
    __syncthreads();
    for (int off = NTHR / 2; off > 0; off >>= 1) {
        if (tid < off) {
            sred_f[tid] += sred_f[tid + off];
            sred_i[tid] += sred_i[tid + off];
        }
        __syncthreads();
    }
    float tsum = 0.0f;
    int   tcnt = 0;
    if (tid == 0) { tsum = sred_f[0]; tcnt = sred_i[0]; }
    __syncthreads();

    // max reduction over the whole trace
    sred_f[tid] = vmax_local;
    __syncthreads();
    for (int off = NTHR / 2; off > 0; off >>= 1) {
        if (tid < off) sred_f[tid] = fmaxf(sred_f[tid], sred_f[tid + off]);
        __syncthreads();
    }

    if (tid == 0) {
        float vmax = sred_f[0];
        bool is_label = (labels[b] == n);
        float c = 0.0f;
        if (num_cluster > 0) {
            if (!is_label) c = tsum / (float)(tcnt > 0 ? tcnt : 1); // false positive
        } else {
            if (is_label)  c = -vmax;                               // miss
        }
        out_spk[r] = (float)num_cluster;
        if (contrib) contrib[r] = c;
        else if (c != 0.0f) atomicAdd(loss_atomic, c);
    }
}

__global__ __launch_bounds__(NTHR) void stca_reduce_kernel(
    const float* __restrict__ contrib, float* __restrict__ out_loss, int n)
{
    __shared__ float s[NTHR];
    int tid = threadIdx.x;
    float acc = 0.0f;
    for (int i = tid; i < n; i += NTHR) acc += contrib[i];   // fixed order
    s[tid] = acc;
    __syncthreads();
    for (int off = NTHR / 2; off > 0; off >>= 1) {
        if (tid < off) s[tid] += s[tid + off];
        __syncthreads();
    }
    if (tid == 0) out_loss[0] = s[0];
}

extern "C" void kernel_launch(void* const* d_in, const int* in_sizes, int n_in,
                              void* d_out, int out_size, void* d_ws, size_t ws_size,
                              hipStream_t stream) {
    const float* vmem   = (const float*)d_in[0];
    // d_in[1] (vlastmem) is unused by the reference forward pass.
    const int*   labels = (const int*)d_in[2];

    const int Bdim  = in_sizes[2];
    const int total = in_sizes[0];                 // B*N*T
    const int Ndim  = total / (Bdim * T_LEN);
    const int R     = Bdim * Ndim;                 // number of traces

    float* out     = (float*)d_out;                // [0]=loss, [1..R]=spike_output
    float* contrib = (float*)d_ws;
    const bool use_ws = (ws_size >= (size_t)R * sizeof(float));

    if (!use_ws) (void)hipMemsetAsync(out, 0, sizeof(float), stream);

    stca_trace_kernel<<<R, NTHR, 0, stream>>>(
        vmem, labels, out + 1, use_ws ? contrib : nullptr, out, Ndim);

    if (use_ws)
        stca_reduce_kernel<<<1, NTHR, 0, stream>>>(contrib, out, R);
}